// GrantGCN_78271484002573
// MI455X (gfx1250) — compile-verified
//
#include <hip/hip_runtime.h>

// GrantGCN on MI455X (gfx1250, wave32):
//   SpMM via edge-parallel f32 atomics (L2-resident, ~10MB total traffic)
//   Dense layers via V_WMMA_F32_16X16X4_F32 (fp32-exact tensor path).
//   Each GEMM wave owns a full 16-row output strip (all n-tiles) so the
//   A operand is loaded once per K-step and shared across 4 (resp. 3) WMMAs.

#define N_NODES   8192
#define N_EDGES   131072
#define X_DIM     32
#define HIDDEN    64
#define N_CLASSES 40
#define K1        (2 * X_DIM)    // 64  : K of layer-1 GEMM
#define K2        (2 * HIDDEN)   // 128 : K of layer-2 GEMM, also gc2 row stride
#define NT1       (HIDDEN / 16)  // 4 n-tiles in layer 1
#define NT2       3              // ceil(40/16) n-tiles in layer 2

typedef __attribute__((ext_vector_type(2))) float v2f;
typedef __attribute__((ext_vector_type(8))) float v8f;

// ---------------------------------------------------------------------------
// Kernel 1: init workspace.
//   gc [N][64] : cols 0..31 = x, cols 32..63 = 0 (accumulator for A@x)
//   gc2[N][128]: cols 64..127 = 0 (accumulator for A@h); cols 0..63 written
//                later by gemm1.
// ---------------------------------------------------------------------------
__global__ __launch_bounds__(256)
void grantgcn_init(const float* __restrict__ x,
                   float* __restrict__ gc, float* __restrict__ gc2) {
  int tid = blockIdx.x * blockDim.x + threadIdx.x;   // 0 .. N_NODES*64-1
  int m = tid >> 6;
  int c = tid & 63;
  gc[tid] = (c < X_DIM) ? x[m * X_DIM + c] : 0.0f;
  gc2[m * K2 + HIDDEN + c] = 0.0f;
}

// ---------------------------------------------------------------------------
// Kernel 2: y1 = A @ x scattered into gc[:,32:64].
// One wave32 per edge; lane = feature column (X_DIM == 32).
// adj[s,d]=w and adj[d,s]+=w  =>  y1[s] += w*x[d]; y1[d] += w*x[s]
// ---------------------------------------------------------------------------
__global__ __launch_bounds__(256)
void grantgcn_spmm1(const int* __restrict__ ei, const float* __restrict__ ew,
                    const float* __restrict__ x, float* __restrict__ gc) {
  int e = blockIdx.x * 8 + (threadIdx.x >> 5);
  int lane = threadIdx.x & 31;
  if (e >= N_EDGES) return;
  int s = ei[e];
  int d = ei[N_EDGES + e];
  float w = ew[e];
  atomicAdd(&gc[s * K1 + X_DIM + lane], w * x[d * X_DIM + lane]);
  atomicAdd(&gc[d * K1 + X_DIM + lane], w * x[s * X_DIM + lane]);
}

// ---------------------------------------------------------------------------
// Kernel 3: h = relu(gc @ W1^T + b1) -> gc2[:,0:64]
// One wave per 16-row strip: 4 accumulators (16x64 out), A loaded once per
// K-step, shared by 4 WMMAs. 512 waves -> 64 blocks of 256.
// A operand: lane holds M = mBase + lane%16, K = k + 2*(lane/16) + v (VGPR v).
// B operand: lane holds N = 16*t + lane%16, same K striping.
// C/D: VGPR r holds (M = mBase + 8*(lane/16) + r, N = 16*t + lane%16).
// ---------------------------------------------------------------------------
__global__ __launch_bounds__(256)
void grantgcn_gemm1(const float* __restrict__ gc, const float* __restrict__ W1,
                    const float* __restrict__ b1, float* __restrict__ gc2) {
  int wave = blockIdx.x * (blockDim.x >> 5) + (threadIdx.x >> 5);  // mTile
  int lane = threadIdx.x & 31;
  int half = lane >> 4;
  int nl   = lane & 15;
  int mBase = wave * 16;

  v8f acc[NT1];
#pragma unroll
  for (int t = 0; t < NT1; ++t) {
    float bias = b1[t * 16 + nl];
#pragma unroll
    for (int r = 0; r < 8; ++r) acc[t][r] = bias;
  }

  const float* arow = gc + (mBase + nl) * K1;   // A row for this lane's M
  const float* brow[NT1];
#pragma unroll
  for (int t = 0; t < NT1; ++t) brow[t] = W1 + (t * 16 + nl) * K1;

#pragma unroll
  for (int k = 0; k < K1; k += 4) {
    int kk = k + 2 * half;
    v2f a;
    a[0] = arow[kk];
    a[1] = arow[kk + 1];
#pragma unroll
    for (int t = 0; t < NT1; ++t) {
      v2f b;
      b[0] = brow[t][kk];
      b[1] = brow[t][kk + 1];
      acc[t] = __builtin_amdgcn_wmma_f32_16x16x4_f32(
          /*neg_a=*/false, a, /*neg_b=*/false, b,
          /*c_mod=*/(short)0, acc[t], /*reuse_a=*/false, /*reuse_b=*/false);
    }
  }

#pragma unroll
  for (int t = 0; t < NT1; ++t) {
#pragma unroll
    for (int r = 0; r < 8; ++r) {
      int row = mBase + 8 * half + r;
      float v = acc[t][r];
      gc2[row * K2 + t * 16 + nl] = v > 0.0f ? v : 0.0f;   // fused ReLU
    }
  }
}

// ---------------------------------------------------------------------------
// Kernel 4: y2 = A @ h scattered into gc2[:,64:128], h read from gc2[:,0:64].
// HIDDEN = 64 -> each lane covers columns {lane, lane+32}.
// ---------------------------------------------------------------------------
__global__ __launch_bounds__(256)
void grantgcn_spmm2(const int* __restrict__ ei, const float* __restrict__ ew,
                    float* __restrict__ gc2) {
  int e = blockIdx.x * 8 + (threadIdx.x >> 5);
  int lane = threadIdx.x & 31;
  if (e >= N_EDGES) return;
  int s = ei[e];
  int d = ei[N_EDGES + e];
  float w = ew[e];
  const float* hs = gc2 + s * K2;        // h[s]
  const float* hd = gc2 + d * K2;        // h[d]
  float* ys = gc2 + s * K2 + HIDDEN;     // y2[s]
  float* yd = gc2 + d * K2 + HIDDEN;     // y2[d]
#pragma unroll
  for (int i = 0; i < 2; ++i) {
    int c = lane + i * 32;
    atomicAdd(&ys[c], w * hd[c]);
    atomicAdd(&yd[c], w * hs[c]);
  }
}

// ---------------------------------------------------------------------------
// Kernel 5: out = gc2 @ W2^T + b2.  M=8192, K=128, N=40 -> 3 n-tiles (48).
// One wave per 16-row strip: 3 accumulators, shared A load per K-step.
// Columns >= 40 clamp their B reads to W2 row 0 (in-bounds) and are masked
// on store; WMMA always runs with full EXEC. 512 waves -> 64 blocks.
// ---------------------------------------------------------------------------
__global__ __launch_bounds__(256)
void grantgcn_gemm2(const float* __restrict__ gc2, const float* __restrict__ W2,
                    const float* __restrict__ b2, float* __restrict__ out) {
  int wave = blockIdx.x * (blockDim.x >> 5) + (threadIdx.x >> 5);  // mTile
  int lane = threadIdx.x & 31;
  int half = lane >> 4;
  int nl   = lane & 15;
  int mBase = wave * 16;

  v8f acc[NT2];
  const float* brow[NT2];
  bool valid[NT2];
#pragma unroll
  for (int t = 0; t < NT2; ++t) {
    int col = t * 16 + nl;
    valid[t] = col < N_CLASSES;
    float bias = valid[t] ? b2[col] : 0.0f;
#pragma unroll
    for (int r = 0; r < 8; ++r) acc[t][r] = bias;
    brow[t] = W2 + (valid[t] ? col : 0) * K2;   // in-bounds for all lanes
  }

  const float* arow = gc2 + (mBase + nl) * K2;

#pragma unroll
  for (int k = 0; k < K2; k += 4) {
    int kk = k + 2 * half;
    v2f a;
    a[0] = arow[kk];
    a[1] = arow[kk + 1];
#pragma unroll
    for (int t = 0; t < NT2; ++t) {
      v2f b;
      b[0] = brow[t][kk];
      b[1] = brow[t][kk + 1];
      acc[t] = __builtin_amdgcn_wmma_f32_16x16x4_f32(
          false, a, false, b, (short)0, acc[t], false, false);
    }
  }

#pragma unroll
  for (int t = 0; t < NT2; ++t) {
    if (valid[t]) {
#pragma unroll
      for (int r = 0; r < 8; ++r) {
        int row = mBase + 8 * half + r;
        out[row * N_CLASSES + t * 16 + nl] = acc[t][r];
      }
    }
  }
}

// ---------------------------------------------------------------------------
extern "C" void kernel_launch(void* const* d_in, const int* in_sizes, int n_in,
                              void* d_out, int out_size, void* d_ws, size_t ws_size,
                              hipStream_t stream) {
  const float* x  = (const float*)d_in[0];   // [8192, 32]
  const int*   ei = (const int*)  d_in[1];   // [2, 131072]
  const float* ew = (const float*)d_in[2];   // [131072]
  // d_in[3] = k (== 2, baked in)
  const float* W1 = (const float*)d_in[4];   // [64, 64]
  const float* b1 = (const float*)d_in[5];   // [64]
  const float* W2 = (const float*)d_in[6];   // [40, 128]
  const float* b2 = (const float*)d_in[7];   // [40]
  float* out = (float*)d_out;                // [8192, 40]

  float* gc  = (float*)d_ws;                 // [8192, 64]   (2 MB)
  float* gc2 = gc + (size_t)N_NODES * K1;    // [8192, 128]  (4 MB)

  // 1) gc[:,0:32] = x, gc[:,32:64] = 0, gc2[:,64:128] = 0
  grantgcn_init<<<(N_NODES * 64) / 256, 256, 0, stream>>>(x, gc, gc2);
  // 2) gc[:,32:64] += A @ x   (edge scatter, 1 wave/edge)
  grantgcn_spmm1<<<N_EDGES / 8, 256, 0, stream>>>(ei, ew, x, gc);
  // 3) gc2[:,0:64] = relu(gc @ W1^T + b1)   (WMMA, 512 waves, 4 tiles/wave)
  grantgcn_gemm1<<<512 / 8, 256, 0, stream>>>(gc, W1, b1, gc2);
  // 4) gc2[:,64:128] += A @ h (edge scatter)
  grantgcn_spmm2<<<N_EDGES / 8, 256, 0, stream>>>(ei, ew, gc2);
  // 5) out = gc2 @ W2^T + b2   (WMMA, 512 waves, 3 tiles/wave)
  grantgcn_gemm2<<<512 / 8, 256, 0, stream>>>(gc2, W2, b2, out);
}